// GCNModel_13804024889633
// MI455X (gfx1250) — compile-verified
//
#include <hip/hip_runtime.h>
#include <hip/hip_bf16.h>

typedef float v2f __attribute__((ext_vector_type(2)));
typedef float v8f __attribute__((ext_vector_type(8)));

#define IN_FEATS 256
#define HIDDEN   256
#define OUT_FEATS 128

// ---------------- degree accumulation ----------------
__global__ void deg_count_kernel(const int* __restrict__ src, const int* __restrict__ dst,
                                 float* __restrict__ degO, float* __restrict__ degI, int E) {
  int e = blockIdx.x * blockDim.x + threadIdx.x;
  if (e < E) {
    atomicAdd(&degO[src[e]], 1.0f);
    atomicAdd(&degI[dst[e]], 1.0f);
  }
}

__global__ void rsqrt_clip_kernel(float* __restrict__ d, int n) {
  int i = blockIdx.x * blockDim.x + threadIdx.x;
  if (i < n) d[i] = rsqrtf(fmaxf(d[i], 1.0f));
}

// ---------------- row-scaled GEMM via f32 WMMA ----------------
// Y[M x NOUT] = (X[M x 256] * rowScale[M, None]) @ W[256 x NOUT]
// block = 256 threads (8 waves), grid.x = ceil(M/16). K fixed at 256.
template <int NOUT>
__global__ __launch_bounds__(256) void gemm_rowscale_wmma(
    const float* __restrict__ X, const float* __restrict__ rowScale,
    const float* __restrict__ W, float* __restrict__ Y, int M) {
  constexpr int K = 256;
  // stride 260: 16B-aligned rows for B128 LDS stores; mod 64 == 4 so the
  // A-fragment column reads (banks 4*l15+k0 / 4*l15+k0+2) stay conflict-free.
  constexpr int LDSS = K + 4;
  __shared__ float Alds[16 * LDSS];

  const int m0 = blockIdx.x * 16;
  const int tid = threadIdx.x;
  const int lane = tid & 31;
  const int wave = tid >> 5;

  // Cooperative vectorized load of the 16x256 A stripe, scaled by rsqrt(deg_out).
  // 16 rows * 64 float4 = 1024 vectors; 256 threads -> 4 each.
#pragma unroll
  for (int i = 0; i < 4; ++i) {
    int flat = i * 256 + tid;   // float4 index within stripe
    int row = flat >> 6;        // 64 float4 per row
    int col4 = flat & 63;
    int gr = m0 + row;
    int grc = gr < M ? gr : (M - 1);
    float s = rowScale[grc];
    float4 x = *(const float4*)(X + (size_t)grc * K + col4 * 4);
    float4 v = make_float4(x.x * s, x.y * s, x.z * s, x.w * s);
    *(float4*)&Alds[row * LDSS + col4 * 4] = v;
  }
  __syncthreads();

  const int half = lane >> 4;  // 0: lanes 0-15, 1: lanes 16-31
  const int l15 = lane & 15;
  const bool fullTile = (m0 + 16 <= M);

  for (int nt = wave; nt < NOUT / 16; nt += 8) {
    const int n0 = nt * 16;
    v8f c = {};
    // A-frag: row = l15, base K offset 2*half (ISA 16x4 f32 A layout)
    const float* arow = &Alds[l15 * LDSS + 2 * half];
    // B-frag: N = n0 + l15, K rows k0+2*half and k0+2*half+1
    const float* wbase = W + (size_t)(2 * half) * NOUT + n0 + l15;

#pragma unroll 8
    for (int k0 = 0; k0 < K; k0 += 4) {
      v2f a, b;
      a[0] = arow[k0];
      a[1] = arow[k0 + 1];
      const float* wp = wbase + (size_t)k0 * NOUT;
      b[0] = wp[0];
      b[1] = wp[NOUT];
      c = __builtin_amdgcn_wmma_f32_16x16x4_f32(false, a, false, b, (short)0, c,
                                                false, false);
    }

    // D layout: VGPR r -> (M = r + 8*half, N = l15)
    const int rowb = m0 + 8 * half;
    float* yp = Y + (size_t)rowb * NOUT + n0 + l15;
    if (fullTile) {
#pragma unroll
      for (int r = 0; r < 8; ++r) yp[(size_t)r * NOUT] = c[r];
    } else {
#pragma unroll
      for (int r = 0; r < 8; ++r)
        if (rowb + r < M) yp[(size_t)r * NOUT] = c[r];
    }
  }
}

// ---------------- SpMM scatter: AGG[dst] += H[src] (float4 gather) ----------
template <int D>
__global__ __launch_bounds__(256) void spmm_scatter4(
    const float* __restrict__ H, const int* __restrict__ src,
    const int* __restrict__ dst, float* __restrict__ AGG, int totalVec) {
  constexpr int DV = D / 4;
  int idx = blockIdx.x * blockDim.x + threadIdx.x;
  if (idx >= totalVec) return;
  int e = idx / DV;           // power of two -> shift
  int cv = idx - e * DV;
  const float4 v = *(const float4*)(H + (size_t)src[e] * D + cv * 4);
  float* ap = AGG + (size_t)dst[e] * D + cv * 4;
  atomicAdd(ap + 0, v.x);
  atomicAdd(ap + 1, v.y);
  atomicAdd(ap + 2, v.z);
  atomicAdd(ap + 3, v.w);
}

// ---------------- normalize + bias + relu (float4) ----------------
template <int D>
__global__ void norm_bias_relu4(const float* __restrict__ AGG,
                                const float* __restrict__ degI,
                                const float* __restrict__ bias,
                                float* __restrict__ OUT, int totalVec) {
  constexpr int DV = D / 4;
  int idx = blockIdx.x * blockDim.x + threadIdx.x;
  if (idx >= totalVec) return;
  int row = idx / DV;
  int cv = idx - row * DV;
  float s = degI[row];
  float4 a = *(const float4*)(AGG + (size_t)idx * 4);
  float4 b = *(const float4*)(bias + cv * 4);
  float4 o;
  o.x = fmaxf(a.x * s + b.x, 0.0f);
  o.y = fmaxf(a.y * s + b.y, 0.0f);
  o.z = fmaxf(a.z * s + b.z, 0.0f);
  o.w = fmaxf(a.w * s + b.w, 0.0f);
  *(float4*)(OUT + (size_t)idx * 4) = o;
}

// ---------------- per-node dot products with Wp halves ----------------
// one wave32 per node; one float4 per lane covers all 128 columns.
__global__ __launch_bounds__(256) void node_score_kernel(
    const float* __restrict__ X2, const float* __restrict__ Wp,
    float* __restrict__ sS, float* __restrict__ sD, int M) {
  int node = (blockIdx.x * blockDim.x + threadIdx.x) >> 5;
  int lane = threadIdx.x & 31;
  if (node >= M) return;
  int c4 = lane * 4;
  float4 x = *(const float4*)(X2 + (size_t)node * OUT_FEATS + c4);
  float4 wa = *(const float4*)(Wp + c4);
  float4 wb = *(const float4*)(Wp + OUT_FEATS + c4);
  float a = x.x * wa.x + x.y * wa.y + x.z * wa.z + x.w * wa.w;
  float b = x.x * wb.x + x.y * wb.y + x.z * wb.z + x.w * wb.w;
#pragma unroll
  for (int off = 16; off > 0; off >>= 1) {
    a += __shfl_down(a, off, 32);
    b += __shfl_down(b, off, 32);
  }
  if (lane == 0) {
    sS[node] = a;
    sD[node] = b;
  }
}

// ---------------- per-edge sigmoid score ----------------
__global__ void edge_score_kernel(const float* __restrict__ sS,
                                  const float* __restrict__ sD,
                                  const int* __restrict__ src,
                                  const int* __restrict__ dst,
                                  const float* __restrict__ bp,
                                  float* __restrict__ out, int E) {
  int e = blockIdx.x * blockDim.x + threadIdx.x;
  if (e < E) {
    float s = sS[src[e]] + sD[dst[e]] + bp[0];
    out[e] = 1.0f / (1.0f + expf(-s));
  }
}

extern "C" void kernel_launch(void* const* d_in, const int* in_sizes, int n_in,
                              void* d_out, int out_size, void* d_ws, size_t ws_size,
                              hipStream_t stream) {
  const float* features = (const float*)d_in[0];
  const int* src        = (const int*)d_in[1];
  const int* dst        = (const int*)d_in[2];
  // d_in[3] = edge_type (unused by the reference computation)
  const float* W1 = (const float*)d_in[4];
  const float* b1 = (const float*)d_in[5];
  const float* W2 = (const float*)d_in[6];
  const float* b2 = (const float*)d_in[7];
  const float* Wp = (const float*)d_in[8];
  const float* bp = (const float*)d_in[9];
  float* out = (float*)d_out;

  const int N = in_sizes[0] / IN_FEATS;  // 50000
  const int E = in_sizes[1];             // 800000
  const size_t Ns = (size_t)N;

  // Workspace layout (floats): 4N scalars + N*256 (bufA) + N*256 (bufB) = 516N
  float* ws   = (float*)d_ws;
  float* degO = ws;              // N  : rsqrt(clip(deg_out))
  float* degI = degO + Ns;       // N  : rsqrt(clip(deg_in))
  float* sS   = degI + Ns;       // N
  float* sD   = sS + Ns;         // N
  float* bufA = sD + Ns;         // N*256 : H1, then X1 (layer-1 output)
  float* bufB = bufA + Ns * 256; // N*256 : AGG1; reused as H2 | AGG2
  float* h2   = bufB;            // N*128
  float* agg2 = bufB + Ns * 128; // N*128
  float* x2   = bufB;            // N*128 (overwrites H2 once AGG2 is ready)

  const int TB = 256;

  // ---- degrees (zero, count, rsqrt-clip; degO/degI are contiguous) ----
  hipMemsetAsync(degO, 0, 2 * Ns * sizeof(float), stream);
  deg_count_kernel<<<(E + TB - 1) / TB, TB, 0, stream>>>(src, dst, degO, degI, E);
  rsqrt_clip_kernel<<<(2 * N + TB - 1) / TB, TB, 0, stream>>>(degO, 2 * N);

  // ---- layer 1: GEMM -> scatter -> norm+bias+relu ----
  hipMemsetAsync(bufB, 0, Ns * 256 * sizeof(float), stream);  // AGG1 = 0
  gemm_rowscale_wmma<HIDDEN><<<(N + 15) / 16, TB, 0, stream>>>(features, degO, W1,
                                                               bufA, N);
  {
    int totalVec = E * (HIDDEN / 4);
    spmm_scatter4<HIDDEN><<<(totalVec + TB - 1) / TB, TB, 0, stream>>>(bufA, src, dst,
                                                                       bufB, totalVec);
  }
  {
    int totalVec = N * (HIDDEN / 4);
    norm_bias_relu4<HIDDEN><<<(totalVec + TB - 1) / TB, TB, 0, stream>>>(bufB, degI, b1,
                                                                         bufA, totalVec);
  }

  // ---- layer 2: GEMM -> scatter -> norm+bias+relu ----
  hipMemsetAsync(agg2, 0, Ns * 128 * sizeof(float), stream);  // AGG2 = 0
  gemm_rowscale_wmma<OUT_FEATS><<<(N + 15) / 16, TB, 0, stream>>>(bufA, degO, W2,
                                                                  h2, N);
  {
    int totalVec = E * (OUT_FEATS / 4);
    spmm_scatter4<OUT_FEATS><<<(totalVec + TB - 1) / TB, TB, 0, stream>>>(h2, src, dst,
                                                                          agg2, totalVec);
  }
  {
    int totalVec = N * (OUT_FEATS / 4);
    norm_bias_relu4<OUT_FEATS><<<(totalVec + TB - 1) / TB, TB, 0, stream>>>(agg2, degI,
                                                                            b2, x2, totalVec);
  }

  // ---- edge scoring ----
  node_score_kernel<<<(N * 32 + TB - 1) / TB, TB, 0, stream>>>(x2, Wp, sS, sD, N);
  edge_score_kernel<<<(E + TB - 1) / TB, TB, 0, stream>>>(sS, sD, src, dst, bp, out, E);
}